// MaxFeatureFusion_14259291423428
// MI455X (gfx1250) — compile-verified
//
#include <hip/hip_runtime.h>

// Elementwise max of 4 float32 tensors, N = 32*256*64*64 = 33,554,432.
// HBM-bandwidth bound: 640 MiB traffic -> ~29 us at 23.3 TB/s.
// Strategy: b128 non-temporal loads/stores (read-once/write-once streams that
// exceed the 192 MB L2), grid-stride wave32 loop, max3+max VALU folding.

typedef float v4f __attribute__((ext_vector_type(4)));

__global__ __launch_bounds__(256)
void MaxFeatureFusion_14259291423428_kernel(const float* __restrict__ f1,
                                            const float* __restrict__ f2,
                                            const float* __restrict__ f3,
                                            const float* __restrict__ f4,
                                            float* __restrict__ out,
                                            long long n) {
    const long long n4     = n >> 2;                  // number of float4 packets
    const long long tid    = (long long)blockIdx.x * blockDim.x + threadIdx.x;
    const long long stride = (long long)gridDim.x * blockDim.x;

    const v4f* __restrict__ A = (const v4f*)f1;
    const v4f* __restrict__ B = (const v4f*)f2;
    const v4f* __restrict__ C = (const v4f*)f3;
    const v4f* __restrict__ D = (const v4f*)f4;
    v4f* __restrict__       O = (v4f*)out;

    // Vectorized main loop: 4x global_load_b128 (TH=NT) + global_store_b128 (TH=NT)
    for (long long i = tid; i < n4; i += stride) {
        v4f a = __builtin_nontemporal_load(&A[i]);
        v4f b = __builtin_nontemporal_load(&B[i]);
        v4f c = __builtin_nontemporal_load(&C[i]);
        v4f d = __builtin_nontemporal_load(&D[i]);

        v4f r;
#pragma unroll
        for (int k = 0; k < 4; ++k) {
            // folds to v_max3_num_f32 + v_max_num_f32
            r[k] = fmaxf(fmaxf(a[k], b[k]), fmaxf(c[k], d[k]));
        }
        __builtin_nontemporal_store(r, &O[i]);
    }

    // Scalar tail (dead for this shape since N % 4 == 0, kept for safety).
    for (long long i = (n4 << 2) + tid; i < n; i += stride) {
        float r = fmaxf(fmaxf(f1[i], f2[i]), fmaxf(f3[i], f4[i]));
        __builtin_nontemporal_store(r, &out[i]);
    }
}

extern "C" void kernel_launch(void* const* d_in, const int* in_sizes, int n_in,
                              void* d_out, int out_size, void* d_ws, size_t ws_size,
                              hipStream_t stream) {
    (void)n_in; (void)d_ws; (void)ws_size;

    const float* f1 = (const float*)d_in[0];
    const float* f2 = (const float*)d_in[1];
    const float* f3 = (const float*)d_in[2];
    const float* f4 = (const float*)d_in[3];
    float* out      = (float*)d_out;

    const long long n  = (long long)out_size;      // 33,554,432 elements
    const long long n4 = n >> 2;                   // 8,388,608 float4 packets

    const int block = 256;                         // 8 wave32 waves per block
    // One float4 per thread per input; cap the grid so the grid-stride loop
    // covers any size while keeping tens of thousands of waves resident.
    long long want = (n4 + block - 1) / block;
    int grid = (int)(want > 65536 ? 65536 : (want < 1 ? 1 : want));

    MaxFeatureFusion_14259291423428_kernel<<<grid, block, 0, stream>>>(
        f1, f2, f3, f4, out, n);
}